// RGCNPredictor_18846316495153
// MI455X (gfx1250) — compile-verified
//
#include <hip/hip_runtime.h>

typedef __attribute__((ext_vector_type(2))) float v2f;
typedef __attribute__((ext_vector_type(4))) float v4f;
typedef __attribute__((ext_vector_type(8))) float v8f;

#define N_NODES 50000
#define N_REL   8
#define HID     128
#define IN_DIM  16
#define N_EDGES 1000000

// ---------------------------------------------------------------- utilities
__global__ void zero_b32(float* __restrict__ p, long long n) {
  long long i = (long long)blockIdx.x * blockDim.x + threadIdx.x;
  long long stride = (long long)gridDim.x * blockDim.x;
  for (; i < n; i += stride) p[i] = 0.0f;
}

// one thread per edge: count edges per (dst, rel) segment
__global__ void count_edges(const int* __restrict__ dst,
                            const int* __restrict__ et,
                            int* __restrict__ cnt, int E) {
  int e = blockIdx.x * blockDim.x + threadIdx.x;
  if (e < E) atomicAdd(&cnt[dst[e] * N_REL + et[e]], 1);
}

// transpose weights [R, K, N] -> [R, N, K] so the WMMA B-fragment
// ({B[k][n], B[k+1][n]} per lane) becomes a single aligned b64 load.
__global__ void transpose_w(const float* __restrict__ W, float* __restrict__ Wt,
                            int R, int K, int N) {
  long long i = (long long)blockIdx.x * blockDim.x + threadIdx.x;
  if (i >= (long long)R * K * N) return;
  int n = (int)(i % N);
  int k = (int)((i / N) % K);
  int r = (int)(i / ((long long)K * N));
  Wt[((size_t)r * N + n) * K + k] = W[i];
}

// scatter-add features of x[src] into S[(dst,rel)]; 4 features per thread
template <int F>
__global__ void scatter_add(const int* __restrict__ src,
                            const int* __restrict__ dst,
                            const int* __restrict__ et,
                            const float* __restrict__ x,
                            float* __restrict__ S, int E) {
  const int CH = F / 4;
  long long i = (long long)blockIdx.x * blockDim.x + threadIdx.x;
  if (i >= (long long)E * CH) return;
  int e = (int)(i / CH);
  int c = (int)(i % CH);
  int s = src[e], d = dst[e], r = et[e];
  v4f xv = *(const v4f*)(x + (size_t)s * F + (size_t)c * 4);
  float* o = S + ((size_t)d * N_REL + r) * F + (size_t)c * 4;
  atomicAdd(o + 0, xv.x);
  atomicAdd(o + 1, xv.y);
  atomicAdd(o + 2, xv.z);
  atomicAdd(o + 3, xv.w);
}

// divide each (node, rel) segment by max(cnt, 1)
template <int F>
__global__ void normalize_seg(float* __restrict__ S,
                              const int* __restrict__ cnt, int nseg) {
  int s = blockIdx.x * blockDim.x + threadIdx.x;
  if (s >= nseg) return;
  int c = cnt[s];
  if (c > 1) {
    float scale = 1.0f / (float)c;
    float* p = S + (size_t)s * F;
#pragma unroll
    for (int f = 0; f < F; f += 4) {
      v4f v = *(v4f*)(p + f);
      v.x *= scale; v.y *= scale; v.z *= scale; v.w *= scale;
      *(v4f*)(p + f) = v;
    }
  }
}

// ---------------------------------------------------------------- WMMA GEMM
// out[M,128] = Aself[M,F] @ Wself + sum_r Srel[M,r,F] @ Wrel[r] + bias (opt ReLU)
// Weights are pre-transposed: WselfT[n][k], WrelT[r][n][k].
// Wave32: each wave computes a 16(M)x128(N) tile via V_WMMA_F32_16X16X4_F32
// (full fp32 matrix pipe -> exact fp32 like the reference).
// A frag (16x4): lane L -> row L%16, K pair 2*(L/16)+{0,1}  (one b64 load)
// B frag (4x16): lane L -> col L%16, K pair 2*(L/16)+{0,1}  (one b64 load)
// C/D (16x16):   vgpr v, lane L -> row v + 8*(L/16), col L%16
template <int F, bool RELU>
__global__ void rgcn_gemm(const float* __restrict__ Aself,
                          const float* __restrict__ Srel,
                          const float* __restrict__ WselfT,
                          const float* __restrict__ WrelT,
                          const float* __restrict__ bias,
                          float* __restrict__ out) {
  const int wavesPerBlock = blockDim.x >> 5;
  int wave = threadIdx.x >> 5;
  int lane = threadIdx.x & 31;
  int tileM = blockIdx.x * wavesPerBlock + wave;
  if (tileM * 16 >= N_NODES) return;  // wave-uniform: EXEC all-ones below
  int row0 = tileM * 16;
  int mr = lane & 15;  // A row within tile / B column within tile
  int hi = lane >> 4;  // K-pair selector
  int kp = 2 * hi;     // 0 or 2

  v8f acc[8] = {};

  // self-loop block: Aself @ Wself
  {
    const float* arow = Aself + (size_t)(row0 + mr) * F + kp;
    const float* wcol = WselfT + (size_t)mr * F + kp;  // + t*16*F per N-tile
    for (int k0 = 0; k0 < F; k0 += 4) {
      v2f a = *(const v2f*)(arow + k0);
#pragma unroll
      for (int t = 0; t < 8; ++t) {
        v2f b = *(const v2f*)(wcol + (size_t)t * 16 * F + k0);
        acc[t] = __builtin_amdgcn_wmma_f32_16x16x4_f32(
            false, a, false, b, (short)0, acc[t], false, false);
      }
    }
  }
  // per-relation blocks: Srel[:, r, :] @ Wrel[r]
  for (int r = 0; r < N_REL; ++r) {
    const float* arow = Srel + ((size_t)(row0 + mr) * N_REL + r) * F + kp;
    const float* wcol = WrelT + ((size_t)r * HID + mr) * F + kp;
    for (int k0 = 0; k0 < F; k0 += 4) {
      v2f a = *(const v2f*)(arow + k0);
#pragma unroll
      for (int t = 0; t < 8; ++t) {
        v2f b = *(const v2f*)(wcol + (size_t)t * 16 * F + k0);
        acc[t] = __builtin_amdgcn_wmma_f32_16x16x4_f32(
            false, a, false, b, (short)0, acc[t], false, false);
      }
    }
  }
  // epilogue: bias (+ ReLU), store
#pragma unroll
  for (int t = 0; t < 8; ++t) {
    int col = t * 16 + mr;
    float bb = bias[col];
#pragma unroll
    for (int v = 0; v < 8; ++v) {
      float c = acc[t][v] + bb;
      if (RELU) c = fmaxf(c, 0.0f);
      out[(size_t)(row0 + v + 8 * hi) * HID + col] = c;
    }
  }
}

// ---------------------------------------------------------------- launcher
extern "C" void kernel_launch(void* const* d_in, const int* in_sizes, int n_in,
                              void* d_out, int out_size, void* d_ws, size_t ws_size,
                              hipStream_t stream) {
  const int*   edge_index = (const int*)d_in[0];   // [2, E]
  const int*   edge_type  = (const int*)d_in[1];   // [E]
  const float* x     = (const float*)d_in[2];      // [N, 16]
  const float* W1    = (const float*)d_in[3];      // [8, 16, 128]
  const float* root1 = (const float*)d_in[4];      // [16, 128]
  const float* b1    = (const float*)d_in[5];      // [128]
  const float* W2    = (const float*)d_in[6];      // [8, 128, 128]
  const float* root2 = (const float*)d_in[7];      // [128, 128]
  const float* b2    = (const float*)d_in[8];      // [128]
  float* out = (float*)d_out;

  const int* src = edge_index;
  const int* dst = edge_index + N_EDGES;

  // workspace layout (256B aligned chunks)
  char* ws = (char*)d_ws;
  size_t off = 0;
  auto alloc = [&](size_t bytes) {
    char* p = ws + off;
    off = (off + bytes + 255) & ~(size_t)255;
    return p;
  };
  int*   cnt = (int*)  alloc((size_t)N_NODES * N_REL * sizeof(int));
  float* S1  = (float*)alloc((size_t)N_NODES * N_REL * IN_DIM * sizeof(float));
  float* h1  = (float*)alloc((size_t)N_NODES * HID * sizeof(float));
  float* S2  = (float*)alloc((size_t)N_NODES * N_REL * HID * sizeof(float));
  float* W1t = (float*)alloc((size_t)N_REL * IN_DIM * HID * sizeof(float));
  float* r1t = (float*)alloc((size_t)IN_DIM * HID * sizeof(float));
  float* W2t = (float*)alloc((size_t)N_REL * HID * HID * sizeof(float));
  float* r2t = (float*)alloc((size_t)HID * HID * sizeof(float));

  const int T = 256;
  const int NSEG = N_NODES * N_REL;

  // ---- weight transposes (tiny; L2-resident afterwards)
  transpose_w<<<(N_REL * IN_DIM * HID + T - 1) / T, T, 0, stream>>>(
      W1, W1t, N_REL, IN_DIM, HID);
  transpose_w<<<(IN_DIM * HID + T - 1) / T, T, 0, stream>>>(
      root1, r1t, 1, IN_DIM, HID);
  transpose_w<<<(N_REL * HID * HID + T - 1) / T, T, 0, stream>>>(
      W2, W2t, N_REL, HID, HID);
  transpose_w<<<(HID * HID + T - 1) / T, T, 0, stream>>>(
      root2, r2t, 1, HID, HID);

  // ---- shared: per-(dst,rel) edge counts (graph is fixed across layers)
  zero_b32<<<(NSEG + T - 1) / T, T, 0, stream>>>((float*)cnt, NSEG);
  count_edges<<<(N_EDGES + T - 1) / T, T, 0, stream>>>(dst, edge_type, cnt, N_EDGES);

  // ---- layer 1
  {
    long long n = (long long)NSEG * IN_DIM;
    zero_b32<<<(int)((n + T - 1) / T), T, 0, stream>>>(S1, n);
    long long sc = (long long)N_EDGES * (IN_DIM / 4);
    scatter_add<IN_DIM><<<(int)((sc + T - 1) / T), T, 0, stream>>>(
        src, dst, edge_type, x, S1, N_EDGES);
    normalize_seg<IN_DIM><<<(NSEG + T - 1) / T, T, 0, stream>>>(S1, cnt, NSEG);
    int tiles = N_NODES / 16;                  // 3125
    rgcn_gemm<IN_DIM, true><<<(tiles + 3) / 4, 128, 0, stream>>>(
        x, S1, r1t, W1t, b1, h1);
  }
  // ---- layer 2
  {
    long long n = (long long)NSEG * HID;
    zero_b32<<<(int)((n + T - 1) / T), T, 0, stream>>>(S2, n);
    long long sc = (long long)N_EDGES * (HID / 4);
    scatter_add<HID><<<(int)((sc + T - 1) / T), T, 0, stream>>>(
        src, dst, edge_type, h1, S2, N_EDGES);
    normalize_seg<HID><<<(NSEG + T - 1) / T, T, 0, stream>>>(S2, cnt, NSEG);
    int tiles = N_NODES / 16;
    rgcn_gemm<HID, false><<<(tiles + 3) / 4, 128, 0, stream>>>(
        h1, S2, r2t, W2t, b2, out);
  }
}